// TransBasicBlockSparse_39049842655653
// MI455X (gfx1250) — compile-verified
//
#include <hip/hip_runtime.h>

typedef __attribute__((ext_vector_type(2))) float v2f;
typedef __attribute__((ext_vector_type(4))) float v4f;
typedef __attribute__((ext_vector_type(8))) float v8f;

#define NSITES 200000
#define CCH    128      // channels
#define KOFF   9        // 3x3 kernel offsets
#define MT     4        // 16-row M tiles per workgroup
#define ROWS   (MT*16)  // 64 sites per workgroup
#define PITCH  132      // LDS row pitch in dwords (128 + 4 pad -> conflict-free A reads)

// Submanifold conv tile kernel.
// MODE 0: out = relu(conv(in))
// MODE 1: out = relu(conv(in) + res)   (residual add)
template<int MODE>
__global__ __launch_bounds__(256)
void subm_conv_wmma(const float* __restrict__ in,
                    const int*   __restrict__ nbr,
                    const float* __restrict__ w,     // [K,128,128]
                    const float* __restrict__ res,   // residual (MODE 1)
                    float* __restrict__ out)
{
    __shared__ float smem[ROWS * PITCH];   // gathered neighbor rows for one k

    const int tid  = threadIdx.x;
    const int wave = tid >> 5;             // 0..7 -> output-channel tile
    const int lane = tid & 31;
    const int lo   = lane & 15;
    const int hi   = lane >> 4;            // half-wave select (K pair / M high rows)
    const int base = blockIdx.x * ROWS;    // first site of this workgroup
    const int n0   = wave * 16;            // first output channel of this wave

    v8f acc[MT] = {};                      // 4 x (16x16 f32 accumulator)

    // gather mapping: 4 consecutive threads stage one 512B feature row
    const int g_r = tid >> 2;              // row 0..63
    const int g_p = tid & 3;               // 32-float slice within row

    for (int k = 0; k < KOFF; ++k) {
        // ---- stage gathered neighbor rows into LDS (coalesced b128 loads) ----
        {
            const int site = base + g_r;
            const int idx  = nbr[site * KOFF + k];
            float* dst = &smem[g_r * PITCH + g_p * 32];
            if (idx < NSITES) {
                const float* src = in + (size_t)idx * CCH + g_p * 32;
                #pragma unroll
                for (int i = 0; i < 8; ++i)
                    ((v4f*)dst)[i] = ((const v4f*)src)[i];
            } else {                        // sentinel neighbor -> zero row
                #pragma unroll
                for (int i = 0; i < 8; ++i)
                    ((v4f*)dst)[i] = (v4f)0.0f;
            }
        }
        __syncthreads();

        // ---- 32 K-chunks of 4 along C_in, fp32 WMMA 16x16x4 ----
        const float* wk = w + (size_t)k * CCH * CCH;
        #pragma unroll 4
        for (int ch = 0; ch < 32; ++ch) {
            const int krow = ch * 4 + 2 * hi;   // this half-wave's K pair
            // B fragment: rows krow/krow+1 of W[k], cols n0..n0+15
            v2f b;
            b.x = wk[(size_t)krow       * CCH + n0 + lo];
            b.y = wk[(size_t)(krow + 1) * CCH + n0 + lo];
            #pragma unroll
            for (int mt = 0; mt < MT; ++mt) {
                // A fragment from LDS: row (mt*16+lo), channels krow..krow+1
                const v2f a = *(const v2f*)&smem[(mt * 16 + lo) * PITCH + krow];
                acc[mt] = __builtin_amdgcn_wmma_f32_16x16x4_f32(
                    /*neg_a=*/false, a, /*neg_b=*/false, b,
                    /*c_mod=*/(short)0, acc[mt],
                    /*reuse_a=*/false, /*reuse_b=*/false);
            }
        }
        __syncthreads();   // all waves done reading before next gather overwrites
    }

    // ---- epilogue: C/D layout is VGPR j, lanes 0-15 -> M=j, lanes 16-31 -> M=8+j ----
    #pragma unroll
    for (int mt = 0; mt < MT; ++mt) {
        #pragma unroll
        for (int j = 0; j < 8; ++j) {
            const int m = mt * 16 + j + 8 * hi;
            const size_t off = (size_t)(base + m) * CCH + n0 + lo;
            float v = acc[mt][j];
            if (MODE == 1) v += res[off];
            out[off] = fmaxf(v, 0.0f);
        }
    }
}

extern "C" void kernel_launch(void* const* d_in, const int* in_sizes, int n_in,
                              void* d_out, int out_size, void* d_ws, size_t ws_size,
                              hipStream_t stream)
{
    const float* feats = (const float*)d_in[0];   // [N,128] f32
    const int*   nbr   = (const int*)  d_in[1];   // [N,9] int32 (sentinel == N)
    const float* W1    = (const float*)d_in[2];   // [9,128,128] f32
    const float* W2    = (const float*)d_in[3];   // [9,128,128] f32
    float* out = (float*)d_out;                   // [N,128] f32
    float* mid = (float*)d_ws;                    // intermediate relu(conv1(x)), N*128 f32

    dim3 grid(NSITES / ROWS);   // 200000 / 64 = 3125 exactly
    dim3 block(256);

    // conv1 + relu -> mid
    subm_conv_wmma<0><<<grid, block, 0, stream>>>(feats, nbr, W1, nullptr, mid);
    // conv2 + residual + relu -> out
    subm_conv_wmma<1><<<grid, block, 0, stream>>>(mid, nbr, W2, feats, out);
}